// GCN_33569464386076
// MI455X (gfx1250) — compile-verified
//
#include <hip/hip_runtime.h>

#define NN 100000           // nodes
#define NE 1600000          // edges
#define FD 128              // feature dim (in & hidden)

typedef __attribute__((ext_vector_type(2))) float v2f;
typedef __attribute__((ext_vector_type(8))) float v8f;

// ---------------------------------------------------------------------------
// Zero the aggregation buffer (float4 stores)
// ---------------------------------------------------------------------------
__global__ void gcn_zero(float4* __restrict__ p, int n4) {
    int i = blockIdx.x * blockDim.x + threadIdx.x;
    if (i < n4) p[i] = make_float4(0.f, 0.f, 0.f, 0.f);
}

// ---------------------------------------------------------------------------
// Wt[n][k] = W[k][n]  (W: [FD x NCOLS] row-major -> Wt: [NCOLS x FD]).
// One-time tiny kernel per layer so the GEMM B-fragment is a contiguous v2f.
// ---------------------------------------------------------------------------
template <int NCOLS>
__global__ void gcn_transpose_w(const float* __restrict__ W, float* __restrict__ Wt) {
    int i = blockIdx.x * blockDim.x + threadIdx.x;
    if (i < FD * NCOLS) {
        int k = i / NCOLS;
        int n = i - k * NCOLS;
        Wt[n * FD + k] = W[i];
    }
}

// ---------------------------------------------------------------------------
// Edge scatter-add: one wave32 per edge; each lane handles 4 contiguous floats.
// Gather is a coalesced 512B read per wave.  Adds use relaxed, agent-scope
// atomic fadd so they lower to single non-returning global_atomic_add_f32
// (STOREcnt-tracked, no return latency, no fences) resolving in the 192MB L2
// where both x (51.2MB) and agg (51.2MB) are resident.
// ---------------------------------------------------------------------------
__device__ __forceinline__ void fadd_relaxed(float* p, float v) {
    __hip_atomic_fetch_add(p, v, __ATOMIC_RELAXED, __HIP_MEMORY_SCOPE_AGENT);
}

__global__ __launch_bounds__(256) void gcn_scatter(
    const float* __restrict__ x, const int* __restrict__ src,
    const int* __restrict__ dst, float* __restrict__ agg, int n_edges)
{
    int t    = blockIdx.x * blockDim.x + threadIdx.x;
    int edge = t >> 5;
    int lane = t & 31;
    if (edge >= n_edges) return;
    int s = src[edge];           // same address across wave -> broadcast
    int d = dst[edge];
    const float4 v = *(const float4*)(x + (size_t)s * FD + lane * 4);
    float* base = agg + (size_t)d * FD + lane * 4;
    fadd_relaxed(base + 0, v.x);
    fadd_relaxed(base + 1, v.y);
    fadd_relaxed(base + 2, v.z);
    fadd_relaxed(base + 3, v.w);
}

// ---------------------------------------------------------------------------
// out = relu(A[NN x FD] @ W[FD x NCOLS] + b), one 16x16 tile per wave via
// V_WMMA_F32_16X16X4_F32, K unrolled FD/4 = 32 WMMAs per tile.
//
// A-frag (16x4 f32, ISA 7.12.2): lanes 0-15 hold {K=k,k+1}, lanes 16-31 hold
// {K=k+2,k+3}, M = lane&15.  B-frag mirrors with N = lane&15 (read from the
// transposed Wt so it's a contiguous 8B load).  C/D: VGPR v holds
// (M = v + 8*(lane>>4), N = lane&15).
//
// NCOLS is a template constant: all strides fold into immediate offsets.
// Out-of-range columns (NCOLS=40 tail tile) are handled by CLAMPING the
// B address (always in-bounds) and guarding only the final store — no
// exec-mask churn inside the WMMA loop.
// ---------------------------------------------------------------------------
template <int NCOLS>
__global__ __launch_bounds__(256) void gcn_gemm_bias_relu(
    const float* __restrict__ A, const float* __restrict__ Wt,
    const float* __restrict__ bias, float* __restrict__ out)
{
    constexpr int TILES_N = (NCOLS + 15) / 16;
    constexpr int TILES_M = NN / 16;                 // 6250, exact
    const int lane  = threadIdx.x & 31;
    const int wave  = threadIdx.x >> 5;
    const int gwave = blockIdx.x * 8 + wave;
    if (gwave >= TILES_M * TILES_N) return;          // uniform per-wave exit
    const int tile_m = gwave / TILES_N;
    const int tile_n = gwave - tile_m * TILES_N;

    const int half = lane >> 4;                      // 0: K+{0,1}, 1: K+{2,3}
    const int l15  = lane & 15;
    const int col  = tile_n * 16 + l15;
    const int ccol = (NCOLS % 16 == 0) ? col : (col < NCOLS ? col : NCOLS - 1);

    const float* arow = A  + (size_t)(tile_m * 16 + l15) * FD + 2 * half;
    const float* brow = Wt + (size_t)ccol * FD + 2 * half;

    v8f c = {};
#pragma unroll
    for (int kk = 0; kk < FD; kk += 4) {
        v2f a = *(const v2f*)(arow + kk);            // global_load_b64, imm offset
        v2f b = *(const v2f*)(brow + kk);            // global_load_b64, imm offset
        c = __builtin_amdgcn_wmma_f32_16x16x4_f32(
                /*neg_a=*/false, a, /*neg_b=*/false, b,
                /*c_mod=*/(short)0, c, /*reuse_a=*/false, /*reuse_b=*/false);
    }

    if (NCOLS % 16 == 0 || col < NCOLS) {
        const float bv = bias[col];
#pragma unroll
        for (int v = 0; v < 8; ++v) {
            const int row = tile_m * 16 + v + 8 * half;
            float val = c[v] + bv;
            out[(size_t)row * NCOLS + col] = val > 0.f ? val : 0.f;
        }
    }
}

// ---------------------------------------------------------------------------
extern "C" void kernel_launch(void* const* d_in, const int* in_sizes, int n_in,
                              void* d_out, int out_size, void* d_ws, size_t ws_size,
                              hipStream_t stream) {
    const float* feats = (const float*)d_in[0];
    const int*   src   = (const int*)  d_in[1];
    const int*   dst   = (const int*)  d_in[2];
    const float* W1    = (const float*)d_in[3];
    const float* b1    = (const float*)d_in[4];
    const float* W2    = (const float*)d_in[5];
    const float* b2    = (const float*)d_in[6];
    const float* W3    = (const float*)d_in[7];
    const float* b3    = (const float*)d_in[8];

    float* A  = (float*)d_ws;                   // 100000*128 f32 = 51.2 MB
    float* H  = A + (size_t)NN * FD;            // 51.2 MB activations
    float* Wt = H + (size_t)NN * FD;            // 64 KB transposed weights

    const int n4 = NN * FD / 4;                 // 3.2M float4
    const dim3 zgrid((n4 + 255) / 256);
    const dim3 sgrid((NE * 32 + 255) / 256);    // 200000 blocks

    constexpr int GB128 = ((NN / 16) * (128 / 16) + 7) / 8;     // 6250 blocks
    constexpr int GB40  = ((NN / 16) * ((40 + 15) / 16) + 7) / 8;

    // Layer 1: agg(feats) -> H = relu(agg@W1 + b1)
    gcn_zero<<<zgrid, 256, 0, stream>>>((float4*)A, n4);
    gcn_transpose_w<128><<<(FD * 128 + 255) / 256, 256, 0, stream>>>(W1, Wt);
    gcn_scatter<<<sgrid, 256, 0, stream>>>(feats, src, dst, A, NE);
    gcn_gemm_bias_relu<128><<<GB128, 256, 0, stream>>>(A, Wt, b1, H);

    // Layer 2: agg(H) -> H = relu(agg@W2 + b2)  (GEMM reads A, writes H: safe)
    gcn_zero<<<zgrid, 256, 0, stream>>>((float4*)A, n4);
    gcn_transpose_w<128><<<(FD * 128 + 255) / 256, 256, 0, stream>>>(W2, Wt);
    gcn_scatter<<<sgrid, 256, 0, stream>>>(H, src, dst, A, NE);
    gcn_gemm_bias_relu<128><<<GB128, 256, 0, stream>>>(A, Wt, b2, H);

    // Layer 3: agg(H) -> out = relu(agg@W3 + b3), NCOLS=40 (partial N tile)
    gcn_zero<<<zgrid, 256, 0, stream>>>((float4*)A, n4);
    gcn_transpose_w<40><<<(FD * 40 + 255) / 256, 256, 0, stream>>>(W3, Wt);
    gcn_scatter<<<sgrid, 256, 0, stream>>>(H, src, dst, A, NE);
    gcn_gemm_bias_relu<40><<<GB40, 256, 0, stream>>>(A, Wt, b3, (float*)d_out);
}